// ModelAbstractiveFusion_4707284157004
// MI455X (gfx1250) — compile-verified
//
#include <hip/hip_runtime.h>
#include <math.h>

#define VOCAB 50265
#define BB 8
#define LL 128
#define SS 1024
#define NTOK (BB * LL)

typedef __attribute__((ext_vector_type(2))) float v2f;
typedef __attribute__((ext_vector_type(8))) float v8f;

__device__ __forceinline__ void online_update(float x, float& m, float& s) {
    // one exp per element, branch-free-ish online softmax accumulator
    float d = x - m;
    float e = __expf(-fabsf(d));
    if (d > 0.0f) { s = fmaf(s, e, 1.0f); m = x; }
    else          { s = s + e; }
}

__device__ __forceinline__ float nan_to_num_f(float x) {
    if (isnan(x)) return 0.0f;
    if (isinf(x)) return x > 0.0f ? 3.4028234663852886e38f : -3.4028234663852886e38f;
    return x;
}

// ---------------- Kernel A: per-row online log-sum-exp + NLL ----------------
__global__ __launch_bounds__(256) void nll_kernel(const float* __restrict__ logits,
                                                  const int* __restrict__ labels,
                                                  float* __restrict__ nll_out) {
    const int row = blockIdx.x;           // 0..1023  (b*128 + l)
    const int tid = threadIdx.x;
    const float* base = logits + (size_t)row * VOCAB;

    // row base is only 4B aligned: (4*row) mod 16. Scalar prologue to 16B.
    const int p = (4 - (row & 3)) & 3;
    const int nv4 = (VOCAB - p) >> 2;
    const int tailStart = p + (nv4 << 2);

    float m = -INFINITY, s = 0.0f;
    if (tid < p) online_update(base[tid], m, s);

    const float4* b4 = (const float4*)(base + p);   // 16B aligned, b128 loads
    for (int i = tid; i < nv4; i += 256) {
        float4 x = b4[i];
        online_update(x.x, m, s);
        online_update(x.y, m, s);
        online_update(x.z, m, s);
        online_update(x.w, m, s);
    }
    if (tailStart + tid < VOCAB) online_update(base[tailStart + tid], m, s);

    // wave32 (m,s) merge
    #pragma unroll
    for (int off = 16; off > 0; off >>= 1) {
        float m2 = __shfl_xor(m, off, 32);
        float s2 = __shfl_xor(s, off, 32);
        float M  = fmaxf(m, m2);
        s = fmaf(s, __expf(m - M), s2 * __expf(m2 - M));
        m = M;
    }

    __shared__ float smM[8], smS[8];
    if ((tid & 31) == 0) { smM[tid >> 5] = m; smS[tid >> 5] = s; }
    __syncthreads();
    if (tid == 0) {
        float M = smM[0], S = smS[0];
        #pragma unroll
        for (int w = 1; w < 8; ++w) {
            float m2 = smM[w], s2 = smS[w];
            float Mn = fmaxf(M, m2);
            S = fmaf(S, __expf(M - Mn), s2 * __expf(m2 - Mn));
            M = Mn;
        }
        int lab = labels[row];
        float xl = base[lab];
        nll_out[row] = M + logf(S) - xl;   // -(x_label - (max + log sumexp))
    }
}

// ---------------- Kernel B: membership labels[b,l] in cand_ids[b,:] ----------------
__global__ __launch_bounds__(1024) void seen_kernel(const int* __restrict__ labels,
                                                    const int* __restrict__ cand,
                                                    int* __restrict__ seen_out) {
    const int b = blockIdx.x;
    const int tid = threadIdx.x;          // 1024 = 128 tokens * 8 scanners
    __shared__ int cLds[SS];
    __shared__ int seenLds[LL];
    cLds[tid] = cand[b * SS + tid];
    if (tid < LL) seenLds[tid] = 0;
    __syncthreads();

    const int l = tid >> 3;
    const int q = tid & 7;
    const int lab = labels[b * LL + l];
    int hit = 0;
    const int s0 = q * (SS / 8);
    #pragma unroll 8
    for (int j = 0; j < SS / 8; ++j) hit |= (cLds[s0 + j] == lab) ? 1 : 0;
    if (hit) atomicOr(&seenLds[l], 1);    // bitwise OR: order-independent, deterministic
    __syncthreads();
    if (tid < LL) seen_out[b * LL + tid] = seenLds[tid];
}

// ---------------- Kernel C: five scalar reductions, WMMA f32 for the tail ----------------
__global__ __launch_bounds__(1024) void finalize_kernel(const float* __restrict__ nll,
                                                        const int* __restrict__ seen,
                                                        float* __restrict__ out) {
    const int tid = threadIdx.x;          // one thread per token
    const int l = tid & (LL - 1);
    float x  = nll[tid];
    float sn = (float)seen[tid];

    float v[5];
    v[0] = x;                               // untouched sum
    v[1] = sn;                              // seen count
    v[2] = x * sn;                          // seen sum
    v[3] = x - x * sn;                      // new sum
    float wn = (sn > 0.0f) ? 1.0f : 2.0f;   // NEW_TOKENS_WEIGHT on new tokens
    float pw = 1.0f + ((float)(l + 1)) * (1.0f / (float)LL);  // long-seq weight
    v[4] = x * wn * pw;                     // weighted sum

    #pragma unroll
    for (int k = 0; k < 5; ++k)
        #pragma unroll
        for (int off = 16; off > 0; off >>= 1)
            v[k] += __shfl_xor(v[k], off, 32);

    __shared__ float P[32][8];              // [wave][quantity], padded
    if ((tid & 31) == 0) {
        #pragma unroll
        for (int k = 0; k < 5; ++k) P[tid >> 5][k] = v[k];
    }
    __syncthreads();

    float tot[5];
    if (tid < 32) {                         // wave 0, EXEC all-ones at the WMMA
        const int lane = tid;
        #pragma unroll
        for (int k = 0; k < 5; ++k) {
            // B(4x16): column n = lane%16. Lanes 0-15 carry P[n][k] and P[n+16][k]
            // in their two B VGPRs (two distinct K rows); lanes 16-31 carry zeros.
            // With A = all-ones, D[m][n] = sum_k B[k][n] = P[n][k] + P[n+16][k],
            // identical in every row and layout-robust.
            v2f bb;
            bb.x = (lane < 16) ? P[lane][k] : 0.0f;
            bb.y = (lane < 16) ? P[lane + 16][k] : 0.0f;
            v2f aa; aa.x = 1.0f; aa.y = 1.0f;
            v8f cc = {};
            v8f dd = __builtin_amdgcn_wmma_f32_16x16x4_f32(
                false, aa, false, bb, (short)0, cc, false, false);
            float t = dd[0];                // lane L holds colsum[L % 16]
            #pragma unroll
            for (int off = 8; off > 0; off >>= 1)  // each 16-half sums to the total
                t += __shfl_xor(t, off, 32);
            tot[k] = t;
        }
    }

    if (tid == 0) {
        const float n_tok = (float)NTOK;
        float untouched  = tot[0] / n_tok;
        float seen_count = tot[1];
        float new_count  = n_tok - seen_count;
        float seen_sum   = tot[2];
        float new_sum    = tot[3];
        float overall    = tot[4] / n_tok;
        float seen_loss  = (seen_count > 0.0f) ? seen_sum / fmaxf(seen_count, 1.0f) : seen_sum;
        float new_loss   = (new_count  > 0.0f) ? new_sum  / fmaxf(new_count,  1.0f) : new_sum;
        float seen_frac  = 100.0f * seen_count / n_tok;
        float new_frac   = 100.0f * new_count  / n_tok;
        out[0] = nan_to_num_f(overall);
        out[1] = untouched;
        out[2] = nan_to_num_f(seen_loss);
        out[3] = seen_frac;
        out[4] = nan_to_num_f(new_loss);
        out[5] = new_frac;
    }
}

extern "C" void kernel_launch(void* const* d_in, const int* in_sizes, int n_in,
                              void* d_out, int out_size, void* d_ws, size_t ws_size,
                              hipStream_t stream) {
    (void)in_sizes; (void)n_in; (void)out_size; (void)ws_size;
    const float* logits = (const float*)d_in[0];
    const int*   labels = (const int*)d_in[1];
    const int*   cand   = (const int*)d_in[2];
    float* out = (float*)d_out;

    float* nll  = (float*)d_ws;                             // 1024 floats
    int*   seen = (int*)((char*)d_ws + NTOK * sizeof(float)); // 1024 ints

    nll_kernel<<<NTOK, 256, 0, stream>>>(logits, labels, nll);
    seen_kernel<<<BB, 1024, 0, stream>>>(labels, cand, seen);
    finalize_kernel<<<1, 1024, 0, stream>>>(nll, seen, out);
}